// DynamicStatFeaturizer_35673998360738
// MI455X (gfx1250) — compile-verified
//
#include <hip/hip_runtime.h>
#include <stdint.h>

typedef float v2f __attribute__((ext_vector_type(2)));
typedef float v8f __attribute__((ext_vector_type(8)));

#define NB 64
#define NJ 5
#define NGROUP (NB*NJ)            // 320 edge-stat groups
#define EDGE_ROW 4433600u         // 177344 * 25 floats per batch row
#define NODE_ROW 973u
#define OUT_COLS 98u
#define CH 65536u                 // virtual elements per pass block
#define BLK_PER_ROW 60u
#define NPASSBLK (NB*BLK_PER_ROW) // 3840 partial slots

// compile-time per-J constants
constexpr unsigned kM[NJ]        = {1728u, 73728u, 294912u, 3276800u, 128000u};
constexpr unsigned kOffE[NJ]     = {0u, 192u, 8384u, 41152u, 172224u};
constexpr unsigned kIs9[NJ]      = {1u, 1u, 1u, 0u, 0u};
constexpr unsigned kBlkStart[NJ] = {0u, 1u, 3u, 8u, 58u};
constexpr unsigned kBlkCnt[NJ]   = {1u, 2u, 5u, 50u, 2u};

__constant__ unsigned c_blkStart[NJ] = {0u, 1u, 3u, 8u, 58u};
__constant__ unsigned c_blkCnt[NJ]   = {1u, 2u, 5u, 50u, 2u};
__constant__ unsigned c_M[NJ]        = {1728u, 73728u, 294912u, 3276800u, 128000u};
__constant__ unsigned c_xoff[NJ]     = {3u, 67u, 195u, 451u, 963u};
__constant__ unsigned c_xlen[NJ]     = {64u, 128u, 256u, 512u, 10u};

__device__ __forceinline__ unsigned monoKey(float f) {
    unsigned u = __float_as_uint(f);
    return (u & 0x80000000u) ? ~u : (u | 0x80000000u);
}
__device__ __forceinline__ float keyToFloat(unsigned k) {
    unsigned u = (k & 0x80000000u) ? (k & 0x7FFFFFFFu) : ~k;
    return __uint_as_float(u);
}
// virtual element index -> float offset within the group's edge block (compile-time crop select)
template <bool IS9>
__device__ __forceinline__ unsigned memOffset(unsigned v) {
    if (IS9) {
        unsigned q  = (unsigned)(((unsigned long long)v * 954437177ull) >> 33);   // v/9
        unsigned r  = v - q * 9u;
        unsigned rr = (r * 11u) >> 5;                                             // r/3
        unsigned col = 5u * rr + (r - 3u * rr) + 6u;  // {6,7,8,11,12,13,16,17,18}
        return q * 25u + col;
    } else {
        unsigned q = (unsigned)(((unsigned long long)v * 1374389535ull) >> 35);   // v/25
        return q * 25u + (v - q * 25u);
    }
}

// ---------------- Pass 1: WMMA sums + min/max + level-0 (12-bit) histogram ----------------
template <unsigned J>
__global__ __launch_bounds__(256) void pass1_kernel(const float* __restrict__ edge,
                                                    unsigned* __restrict__ hist0,
                                                    unsigned* __restrict__ minkey,
                                                    unsigned* __restrict__ maxkey,
                                                    double* __restrict__ partials) {
    constexpr unsigned M = kM[J];
    constexpr bool IS9 = (kIs9[J] != 0u);
    __shared__ unsigned lhist[4096];
    __shared__ double wsum[8], wsq[8];
    __shared__ unsigned wmin[8], wmax[8];
    unsigned tid = threadIdx.x;
    for (unsigned i = tid; i < 4096u; i += 256u) lhist[i] = 0u;
    __syncthreads();

    unsigned chunk = blockIdx.x, b = blockIdx.y;
    unsigned group = b * NJ + J;
    const float* gbase = edge + (size_t)b * EDGE_ROW + (size_t)kOffE[J] * 25u;
    unsigned startV = chunk * CH;
    unsigned endV = min(M, startV + CH);
    unsigned fullEnd = startV + ((endV - startV) & ~511u);

    v8f csum = {0.f,0.f,0.f,0.f,0.f,0.f,0.f,0.f};
    v8f csq  = {0.f,0.f,0.f,0.f,0.f,0.f,0.f,0.f};
    v2f ones; ones[0] = 1.0f; ones[1] = 1.0f;
    unsigned kmin = 0xFFFFFFFFu, kmax = 0u;

    // ---- unguarded hot loop: 512 elements per block-iter, 2 WMMAs ----
    for (unsigned base = startV; base < fullEnd; base += 512u) {
        unsigned v0 = base + tid;
        unsigned v1 = v0 + 256u;
        float x0 = gbase[memOffset<IS9>(v0)];
        float x1 = gbase[memOffset<IS9>(v1)];
        unsigned k0 = monoKey(x0), k1 = monoKey(x1);
        atomicAdd(&lhist[k0 >> 20], 1u);
        atomicAdd(&lhist[k1 >> 20], 1u);
        kmin = min(kmin, min(k0, k1));
        kmax = max(kmax, max(k0, k1));
        v2f a;  a[0]  = x0;       a[1]  = x1;
        v2f a2; a2[0] = x0 * x0;  a2[1] = x1 * x1;
        // D = A(16x4) * ones(4x16) + C : every D[m][n] accumulates row-sums; wave total = 16x true sum
        csum = __builtin_amdgcn_wmma_f32_16x16x4_f32(false, a,  false, ones, (short)0, csum, false, false);
        csq  = __builtin_amdgcn_wmma_f32_16x16x4_f32(false, a2, false, ones, (short)0, csq,  false, false);
    }
    // ---- tail (only j=0's last 192 elements ever hit this) ----
    float tsum = 0.f, tsq = 0.f;
    for (unsigned v = fullEnd + tid; v < endV; v += 256u) {
        float x = gbase[memOffset<IS9>(v)];
        unsigned k = monoKey(x);
        atomicAdd(&lhist[k >> 20], 1u);
        kmin = min(kmin, k); kmax = max(kmax, k);
        tsum += x; tsq += x * x;
    }

    float lsum = 0.f, lsq = 0.f;
#pragma unroll
    for (int i = 0; i < 8; ++i) { lsum += csum[i]; lsq += csq[i]; }
    for (int off = 16; off > 0; off >>= 1) {
        lsum += __shfl_xor(lsum, off, 32);
        lsq  += __shfl_xor(lsq,  off, 32);
        tsum += __shfl_xor(tsum, off, 32);
        tsq  += __shfl_xor(tsq,  off, 32);
        kmin = min(kmin, (unsigned)__shfl_xor((int)kmin, off, 32));
        kmax = max(kmax, (unsigned)__shfl_xor((int)kmax, off, 32));
    }
    unsigned wave = tid >> 5;
    if ((tid & 31u) == 0u) {
        wsum[wave] = (double)lsum * 0.0625 + (double)tsum;  // wmma part is 16x replicated
        wsq[wave]  = (double)lsq  * 0.0625 + (double)tsq;
        wmin[wave] = kmin; wmax[wave] = kmax;
    }
    __syncthreads();
    if (tid == 0) {
        double S = 0.0, Q = 0.0; unsigned mn = 0xFFFFFFFFu, mx = 0u;
        for (int w = 0; w < 8; ++w) { S += wsum[w]; Q += wsq[w]; mn = min(mn, wmin[w]); mx = max(mx, wmax[w]); }
        unsigned slot = b * BLK_PER_ROW + kBlkStart[J] + chunk;
        partials[2u * slot + 0u] = S;
        partials[2u * slot + 1u] = Q;
        atomicMin(&minkey[group], mn);
        atomicMax(&maxkey[group], mx);
    }
    __syncthreads();
    for (unsigned i = tid; i < 4096u; i += 256u) {
        unsigned h = lhist[i];
        if (h) atomicAdd(&hist0[(size_t)group * 4096u + i], h);
    }
}

// ---------------- Select bin + residual rank from level-0 histogram ----------------
__global__ void selectA_kernel(const unsigned* __restrict__ hist0,
                               unsigned* __restrict__ pre12,
                               unsigned* __restrict__ resid1) {
    unsigned g = blockIdx.x * blockDim.x + threadIdx.x;
    if (g >= NGROUP) return;
    unsigned M = c_M[g % NJ];
    unsigned ranks[6];
#pragma unroll
    for (int qi = 0; qi < 3; ++qi) {
        double pos = 0.25 * (double)(qi + 1) * (double)(M - 1u);
        unsigned lo = (unsigned)pos;
        unsigned hi = (lo + 1u < M) ? lo + 1u : lo;
        ranks[2 * qi] = lo; ranks[2 * qi + 1] = hi;
    }
    const unsigned* h = hist0 + (size_t)g * 4096u;
    unsigned c = 0, done = 0;
    for (unsigned bin = 0; bin < 4096u && done != 0x3Fu; ++bin) {
        unsigned hv = h[bin];
        if (hv) {
            unsigned nc = c + hv;
#pragma unroll
            for (int r = 0; r < 6; ++r)
                if (!((done >> r) & 1u) && ranks[r] < nc) {
                    pre12[g * 6u + r] = bin; resid1[g * 6u + r] = ranks[r] - c; done |= (1u << r);
                }
            c = nc;
        }
    }
}

// ---------------- Pass 2: mid-12-bit histograms of prefix-matching elements ----------------
template <unsigned J>
__global__ __launch_bounds__(256) void pass2_kernel(const float* __restrict__ edge,
                                                    const unsigned* __restrict__ pre12,
                                                    unsigned* __restrict__ hist1) {
    constexpr unsigned M = kM[J];
    constexpr bool IS9 = (kIs9[J] != 0u);
    unsigned chunk = blockIdx.x, b = blockIdx.y;
    unsigned group = b * NJ + J;
    const float* gbase = edge + (size_t)b * EDGE_ROW + (size_t)kOffE[J] * 25u;
    unsigned pr[6];
#pragma unroll
    for (int r = 0; r < 6; ++r) pr[r] = pre12[group * 6u + r];
    unsigned startV = chunk * CH, endV = min(M, startV + CH);
    for (unsigned v = startV + threadIdx.x; v < endV; v += 256u) {
        unsigned k = monoKey(gbase[memOffset<IS9>(v)]);
        unsigned top = k >> 20, mid = (k >> 8) & 0xFFFu;
#pragma unroll
        for (int r = 0; r < 6; ++r)
            if (top == pr[r]) atomicAdd(&hist1[((size_t)group * 6u + r) * 4096u + mid], 1u);
    }
}

__global__ void selectB_kernel(const unsigned* __restrict__ hist1,
                               const unsigned* __restrict__ pre12,
                               const unsigned* __restrict__ resid1,
                               unsigned* __restrict__ pre24,
                               unsigned* __restrict__ resid2) {
    unsigned idx = blockIdx.x * blockDim.x + threadIdx.x;
    if (idx >= NGROUP * 6u) return;
    const unsigned* h = hist1 + (size_t)idx * 4096u;
    unsigned target = resid1[idx], c = 0;
    for (unsigned bin = 0; bin < 4096u; ++bin) {
        unsigned hv = h[bin];
        if (target < c + hv) { pre24[idx] = (pre12[idx] << 12) | bin; resid2[idx] = target - c; return; }
        c += hv;
    }
    pre24[idx] = (pre12[idx] << 12); resid2[idx] = 0;
}

// ---------------- Pass 3: final 8-bit histograms ----------------
template <unsigned J>
__global__ __launch_bounds__(256) void pass3_kernel(const float* __restrict__ edge,
                                                    const unsigned* __restrict__ pre24,
                                                    unsigned* __restrict__ hist2) {
    constexpr unsigned M = kM[J];
    constexpr bool IS9 = (kIs9[J] != 0u);
    unsigned chunk = blockIdx.x, b = blockIdx.y;
    unsigned group = b * NJ + J;
    const float* gbase = edge + (size_t)b * EDGE_ROW + (size_t)kOffE[J] * 25u;
    unsigned pr[6];
#pragma unroll
    for (int r = 0; r < 6; ++r) pr[r] = pre24[group * 6u + r];
    unsigned startV = chunk * CH, endV = min(M, startV + CH);
    for (unsigned v = startV + threadIdx.x; v < endV; v += 256u) {
        unsigned k = monoKey(gbase[memOffset<IS9>(v)]);
        unsigned top = k >> 8;
#pragma unroll
        for (int r = 0; r < 6; ++r)
            if (top == pr[r]) atomicAdd(&hist2[((size_t)group * 6u + r) * 256u + (k & 0xFFu)], 1u);
    }
}

__global__ void selectC_kernel(const unsigned* __restrict__ hist2,
                               const unsigned* __restrict__ pre24,
                               const unsigned* __restrict__ resid2,
                               float* __restrict__ qval) {
    unsigned idx = blockIdx.x * blockDim.x + threadIdx.x;
    if (idx >= NGROUP * 6u) return;
    const unsigned* h = hist2 + (size_t)idx * 256u;
    unsigned target = resid2[idx], c = 0;
    for (unsigned bin = 0; bin < 256u; ++bin) {
        unsigned hv = h[bin];
        if (target < c + hv) { qval[idx] = keyToFloat((pre24[idx] << 8) | bin); return; }
        c += hv;
    }
    qval[idx] = keyToFloat(pre24[idx] << 8);
}

// ---------------- Final assembly of edge-group stats ----------------
__global__ void assemble_kernel(const double* __restrict__ partials,
                                const unsigned* __restrict__ minkey,
                                const unsigned* __restrict__ maxkey,
                                const float* __restrict__ qval,
                                float* __restrict__ out) {
    unsigned g = blockIdx.x * blockDim.x + threadIdx.x;
    if (g >= NGROUP) return;
    unsigned b = g / NJ, j = g % NJ;
    unsigned M = c_M[j];
    double S = 0.0, Q = 0.0;
    unsigned s0 = b * BLK_PER_ROW + c_blkStart[j];
    for (unsigned k = 0; k < c_blkCnt[j]; ++k) { S += partials[2u*(s0+k)]; Q += partials[2u*(s0+k)+1u]; }
    double mean = S / (double)M;
    double var  = (Q - S * S / (double)M) / (double)(M - 1u);
    float* row = out + (size_t)b * OUT_COLS + (size_t)j * 14u;
    row[0] = (float)mean;
    row[1] = (float)var;
    row[2] = keyToFloat(minkey[g]);
#pragma unroll
    for (int qi = 0; qi < 3; ++qi) {
        double pos = 0.25 * (double)(qi + 1) * (double)(M - 1u);
        double lo = floor(pos);
        double frac = pos - lo;
        double vlo = (double)qval[g * 6u + 2 * qi];
        double vhi = (double)qval[g * 6u + 2 * qi + 1];
        row[3 + qi] = (float)(vlo + frac * (vhi - vlo));
    }
    row[6] = keyToFloat(maxkey[g]);
}

// ---------------- Exact stats for tiny x segments: bitonic sort in LDS ----------------
__global__ __launch_bounds__(256) void xstats_kernel(const float* __restrict__ x,
                                                     float* __restrict__ out) {
    __shared__ float s[512];
    __shared__ double red[256];
    unsigned tid = threadIdx.x;
    unsigned g = blockIdx.x;
    unsigned b = g / NJ, j = g % NJ;
    unsigned N = c_xlen[j];
    const float* src = x + (size_t)b * NODE_ROW + c_xoff[j];
    for (unsigned i = tid; i < 512u; i += 256u)
        s[i] = (i < N) ? src[i] : __uint_as_float(0x7F800000u); // +inf pad
    __syncthreads();
    for (unsigned k = 2u; k <= 512u; k <<= 1u)
        for (unsigned jj = k >> 1; jj > 0u; jj >>= 1u) {
            for (unsigned i = tid; i < 512u; i += 256u) {
                unsigned ixj = i ^ jj;
                if (ixj > i) {
                    float a = s[i], c = s[ixj];
                    bool up = ((i & k) == 0u);
                    if (up ? (a > c) : (a < c)) { s[i] = c; s[ixj] = a; }
                }
            }
            __syncthreads();
        }
    double ls = 0.0, lq = 0.0;
    for (unsigned i = tid; i < N; i += 256u) { double v = (double)s[i]; ls += v; lq += v * v; }
    red[tid] = ls; __syncthreads();
    for (unsigned st = 128u; st > 0u; st >>= 1u) { if (tid < st) red[tid] += red[tid + st]; __syncthreads(); }
    double S = red[0]; __syncthreads();
    red[tid] = lq; __syncthreads();
    for (unsigned st = 128u; st > 0u; st >>= 1u) { if (tid < st) red[tid] += red[tid + st]; __syncthreads(); }
    double Q = red[0];
    if (tid == 0) {
        float* row = out + (size_t)b * OUT_COLS + (size_t)j * 14u + 7u;
        double mean = S / (double)N;
        double var  = (Q - S * S / (double)N) / (double)(N - 1u);
        row[0] = (float)mean;
        row[1] = (float)var;
        row[2] = s[0];
        for (int qi = 0; qi < 3; ++qi) {
            double pos = 0.25 * (double)(qi + 1) * (double)(N - 1u);
            unsigned lo = (unsigned)pos;
            double frac = pos - (double)lo;
            unsigned hi = (lo + 1u < N) ? lo + 1u : lo;
            row[3 + qi] = (float)((double)s[lo] + frac * ((double)s[hi] - (double)s[lo]));
        }
        row[6] = s[N - 1u];
    }
}

extern "C" void kernel_launch(void* const* d_in, const int* in_sizes, int n_in,
                              void* d_out, int out_size, void* d_ws, size_t ws_size,
                              hipStream_t stream) {
    (void)in_sizes; (void)n_in; (void)ws_size;
    const float* x    = (const float*)d_in[0];
    const float* edge = (const float*)d_in[1];
    float* out = (float*)d_out;

    uint8_t* ws = (uint8_t*)d_ws;
    size_t off = 0;
    auto alloc = [&](size_t bytes) -> void* {
        void* p = ws + off; off += (bytes + 255u) & ~(size_t)255u; return p;
    };
    unsigned* hist0   = (unsigned*)alloc((size_t)NGROUP * 4096u * 4u);       // 5.2 MB
    unsigned* hist1   = (unsigned*)alloc((size_t)NGROUP * 6u * 4096u * 4u);  // 31.5 MB
    unsigned* hist2   = (unsigned*)alloc((size_t)NGROUP * 6u * 256u * 4u);   // 2.0 MB
    unsigned* minkey  = (unsigned*)alloc((size_t)NGROUP * 4u);
    unsigned* maxkey  = (unsigned*)alloc((size_t)NGROUP * 4u);
    double*   partials= (double*)  alloc((size_t)NPASSBLK * 2u * 8u);
    unsigned* pre12   = (unsigned*)alloc((size_t)NGROUP * 6u * 4u);
    unsigned* resid1  = (unsigned*)alloc((size_t)NGROUP * 6u * 4u);
    unsigned* pre24   = (unsigned*)alloc((size_t)NGROUP * 6u * 4u);
    unsigned* resid2  = (unsigned*)alloc((size_t)NGROUP * 6u * 4u);
    float*    qval    = (float*)   alloc((size_t)NGROUP * 6u * 4u);

    hipMemsetAsync(hist0, 0, (size_t)NGROUP * 4096u * 4u, stream);
    hipMemsetAsync(hist1, 0, (size_t)NGROUP * 6u * 4096u * 4u, stream);
    hipMemsetAsync(hist2, 0, (size_t)NGROUP * 6u * 256u * 4u, stream);
    hipMemsetAsync(minkey, 0xFF, (size_t)NGROUP * 4u, stream);
    hipMemsetAsync(maxkey, 0x00, (size_t)NGROUP * 4u, stream);
    hipMemsetAsync(d_out, 0, (size_t)out_size * sizeof(float), stream);

    xstats_kernel<<<NGROUP, 256, 0, stream>>>(x, out);

    pass1_kernel<0><<<dim3(kBlkCnt[0], NB), 256, 0, stream>>>(edge, hist0, minkey, maxkey, partials);
    pass1_kernel<1><<<dim3(kBlkCnt[1], NB), 256, 0, stream>>>(edge, hist0, minkey, maxkey, partials);
    pass1_kernel<2><<<dim3(kBlkCnt[2], NB), 256, 0, stream>>>(edge, hist0, minkey, maxkey, partials);
    pass1_kernel<3><<<dim3(kBlkCnt[3], NB), 256, 0, stream>>>(edge, hist0, minkey, maxkey, partials);
    pass1_kernel<4><<<dim3(kBlkCnt[4], NB), 256, 0, stream>>>(edge, hist0, minkey, maxkey, partials);

    selectA_kernel<<<(NGROUP + 255) / 256, 256, 0, stream>>>(hist0, pre12, resid1);

    pass2_kernel<0><<<dim3(kBlkCnt[0], NB), 256, 0, stream>>>(edge, pre12, hist1);
    pass2_kernel<1><<<dim3(kBlkCnt[1], NB), 256, 0, stream>>>(edge, pre12, hist1);
    pass2_kernel<2><<<dim3(kBlkCnt[2], NB), 256, 0, stream>>>(edge, pre12, hist1);
    pass2_kernel<3><<<dim3(kBlkCnt[3], NB), 256, 0, stream>>>(edge, pre12, hist1);
    pass2_kernel<4><<<dim3(kBlkCnt[4], NB), 256, 0, stream>>>(edge, pre12, hist1);

    selectB_kernel<<<(NGROUP * 6 + 255) / 256, 256, 0, stream>>>(hist1, pre12, resid1, pre24, resid2);

    pass3_kernel<0><<<dim3(kBlkCnt[0], NB), 256, 0, stream>>>(edge, pre24, hist2);
    pass3_kernel<1><<<dim3(kBlkCnt[1], NB), 256, 0, stream>>>(edge, pre24, hist2);
    pass3_kernel<2><<<dim3(kBlkCnt[2], NB), 256, 0, stream>>>(edge, pre24, hist2);
    pass3_kernel<3><<<dim3(kBlkCnt[3], NB), 256, 0, stream>>>(edge, pre24, hist2);
    pass3_kernel<4><<<dim3(kBlkCnt[4], NB), 256, 0, stream>>>(edge, pre24, hist2);

    selectC_kernel<<<(NGROUP * 6 + 255) / 256, 256, 0, stream>>>(hist2, pre24, resid2, qval);
    assemble_kernel<<<(NGROUP + 255) / 256, 256, 0, stream>>>(partials, minkey, maxkey, qval, out);
}